// BPICDetector__65618510348900
// MI455X (gfx1250) — compile-verified
//
#include <hip/hip_runtime.h>

typedef float v2f __attribute__((ext_vector_type(2)));
typedef float v8f __attribute__((ext_vector_type(8)));

#define BATCH 128
#define SEQ   512
#define NPROB (BATCH * SEQ)   // 65536 independent problems
#define MDIM  16
#define KDIM  8

// Gray-coded 16-QAM point c (unit average energy, 1/sqrt(10) scale).
// pam[j] = (j>>1 ? +1 : -1) * (j&1 ? 1 : 3)  ==  {-3,-1,3,1}
__device__ __forceinline__ float2 cpt(int c) {
  int jr = (((c >> 3) & 1) << 1) | ((c >> 1) & 1);
  int ji = (((c >> 2) & 1) << 1) | (c & 1);
  float re = ((jr >> 1) ? 1.f : -1.f) * ((jr & 1) ? 1.f : 3.f);
  float im = ((ji >> 1) ? 1.f : -1.f) * ((ji & 1) ? 1.f : 3.f);
  return make_float2(re * 0.31622776601683794f, im * 0.31622776601683794f);
}

// Low 32 bits of a generic pointer to LDS == the LDS byte offset
// (aperture check truncates the address to its low 32 bits).
__device__ __forceinline__ unsigned ldsoff(const void* p) {
  return (unsigned)(unsigned long long)p;
}

// gfx1250 async global -> LDS copy, 16B per active lane, ASYNCcnt-tracked.
__device__ __forceinline__ void async_ld_b128(unsigned lds, const void* g) {
  asm volatile("global_load_async_to_lds_b128 %0, %1, off"
               :: "v"(lds), "v"((unsigned long long)g) : "memory");
}

__global__ __launch_bounds__(256) void bpic_wmma_kernel(
    const float* __restrict__ y,      // [NPROB,16] complex64 (interleaved f32)
    const float* __restrict__ h,      // [NPROB,16,8] complex64
    const float* __restrict__ prior,  // [NPROB,8,4] f32
    const float* __restrict__ no,     // 16 f32, scalar lives at [15]
    const int*   __restrict__ itp,    // internal_it (device scalar)
    float* __restrict__ out)          // x_bse | v_bse | ise_dsc, each [NPROB,8] c64
{
  __shared__ __align__(16) float Hs[8][256];  // raw h stage (1 KB/wave)
  __shared__ __align__(16) float Ys[8][33];   // raw y stage + zero pad slot [32]
  __shared__ float Gs[8][256];                // 16x16 Gram spill
  __shared__ float Ts[8][48];                 // Hty cols [0..31] + dump [32..47]

  const int lane = threadIdx.x & 31;
  const int wav  = threadIdx.x >> 5;
  const int p    = blockIdx.x * 8 + wav;      // one wave32 per problem

  const float No    = no[15];                 // no[0,0,1,1,1,1]
  const float rsNo  = rsqrtf(No);
  const int   iters = *itp;

  const float* hp = h     + (size_t)p * (MDIM * KDIM * 2);
  const float* yp = y     + (size_t)p * (MDIM * 2);
  const float* pp = prior + (size_t)p * (KDIM * 4);

  // ---- coalesced async staging: 1 KB of h (2 x B128/lane) + 128 B of y ----
  async_ld_b128(ldsoff(&Hs[wav][lane * 8]),     hp + lane * 8);
  async_ld_b128(ldsoff(&Hs[wav][lane * 8 + 4]), hp + lane * 8 + 4);
  if (lane < 8)
    async_ld_b128(ldsoff(&Ys[wav][lane * 4]), yp + lane * 4);
  Ys[wav][32] = 0.f;                     // shared zero pad (all lanes, same addr)
  asm volatile("s_wait_asynccnt 0x0" ::: "memory");

  const int lh   = lane >> 4;      // lane half (K-pair select in fragments)
  const int l15  = lane & 15;
  const int cc   = l15 & 7;        // column within Re/Im block
  const int isim = l15 >> 3;       // 0 = Re block, 1 = Im block

  // ================= HtH and Hty via WMMA =================
  // X[r][c] = (c<8 ? Re(h[r][c]) : Im(h[r][c-8])) / sqrt(No)   (16x16 real)
  // Y[r][0] = Re(y[r])/sqrt(No), Y[r][1] = Im(y[r])/sqrt(No), else 0
  //   G  = X^T X : HtH blocks     (A-frag == B-frag, lane-for-lane)
  //   G2 = X^T Y : Hty columns
  v8f acc  = {0.f, 0.f, 0.f, 0.f, 0.f, 0.f, 0.f, 0.f};
  v8f acc2 = {0.f, 0.f, 0.f, 0.f, 0.f, 0.f, 0.f, 0.f};
  const float* hb = &Hs[wav][cc * 2 + isim + lh * 32];  // per-lane base, imm offs
  const float* ys = Ys[wav];
  const bool   yok = (l15 < 2);
#pragma unroll
  for (int kk = 0; kk < 4; ++kk) {
    const int r0 = kk * 4 + 2 * lh;
    // branch-free index select: invalid lanes read the zero pad slot [32]
    const int i0 = yok ? (r0 * 2 + l15)       : 32;
    const int i1 = yok ? ((r0 + 1) * 2 + l15) : 32;
    v2f f, f2;
    f.x  = hb[kk * 64]      * rsNo;   // X[r0][l15]
    f.y  = hb[kk * 64 + 16] * rsNo;   // X[r0+1][l15]
    f2.x = ys[i0] * rsNo;             // Y[r0][l15]
    f2.y = ys[i1] * rsNo;             // Y[r0+1][l15]
    acc  = __builtin_amdgcn_wmma_f32_16x16x4_f32(false, f, false, f,
                                                 (short)0, acc,  false, false);
    acc2 = __builtin_amdgcn_wmma_f32_16x16x4_f32(false, f, false, f2,
                                                 (short)0, acc2, false, false);
  }
  // C fragment: VGPR v of lane l holds D[v + 8*lh][l15]
  float* g = Gs[wav];
#pragma unroll
  for (int v = 0; v < 8; ++v) g[(v + 8 * lh) * 16 + l15] = acc[v];
  // Hty spill, branch-free: invalid lanes dump into Ts[32..47]
  {
    const int tb = yok ? (16 * lh + l15) : 32;
#pragma unroll
    for (int v = 0; v < 8; ++v) Ts[wav][tb + v * 2] = acc2[v];
  }
  // Intra-wave producer/consumer only: drain DScnt instead of a block barrier.
  asm volatile("s_wait_dscnt 0x0" ::: "memory");

  // ================= per-symbol setup (k replicated over lane groups) ======
  const int k = lane & 7;

  // Hty from G2:  re = G2[k][0] + G2[k+8][1],  im = G2[k][1] - G2[k+8][0]
  const float tyre = Ts[wav][k * 2]     + Ts[wav][(k + 8) * 2 + 1];
  const float tyim = Ts[wav][k * 2 + 1] - Ts[wav][(k + 8) * 2];

  // complex diagonal of HtH and mrc = 1/diag
  const float dre = g[k * 16 + k] + g[(k + 8) * 16 + (k + 8)];
  const float dim = g[k * 16 + (k + 8)] - g[(k + 8) * 16 + k];
  const float dden = dre * dre + dim * dim;
  const float mre =  dre / dden;           // mrc (complex reciprocal)
  const float mim = -dim / dden;
  const float m2r = mre * mre - mim * mim; // mrc^2
  const float m2i = 2.f * mre * mim;

  // off-diagonal row k of HtH (+ elementwise complex square) in registers
  float hr[8], hi[8], sr[8], si[8];
#pragma unroll
  for (int c = 0; c < 8; ++c) {
    float re = g[k * 16 + c] + g[(k + 8) * 16 + (c + 8)];
    float im = g[k * 16 + (c + 8)] - g[(k + 8) * 16 + c];
    if (c == k) { re = 0.f; im = 0.f; }
    hr[c] = re; hi[c] = im;
    sr[c] = re * re - im * im;
    si[c] = 2.f * re * im;
  }

  // prior LLRs -> symbol moments (softmax over 16 logits)
  const float l0 = pp[k * 4 + 0], l1 = pp[k * 4 + 1];
  const float l2 = pp[k * 4 + 2], l3 = pp[k * 4 + 3];
  float lg[16], lmax = -1e30f;
#pragma unroll
  for (int c = 0; c < 16; ++c) {
    lg[c] = (float)((c >> 3) & 1) * l0 + (float)((c >> 2) & 1) * l1 +
            (float)((c >> 1) & 1) * l2 + (float)(c & 1) * l3;
    lmax = fmaxf(lmax, lg[c]);
  }
  float psum = 0.f, xhr = 0.f, xhi = 0.f, e2 = 0.f;
#pragma unroll
  for (int c = 0; c < 16; ++c) {
    const float w = __expf(lg[c] - lmax);
    const float2 P = cpt(c);
    psum += w;
    xhr += w * P.x; xhi += w * P.y;
    e2  += w * (P.x * P.x + P.y * P.y);
  }
  const float inv = 1.f / psum;
  xhr *= inv; xhi *= inv; e2 *= inv;
  const float var0 = e2 - (xhr * xhr + xhi * xhi);

  // state (replicated across the 4 lanes sharing symbol k)
  float xdr = xhr, xdi = xhi, vdr = var0, vdi = 0.f;       // x_dsc, v_dsc
  float xbr = xhr, xbi = xhi, vor = var0, voi = 0.f;       // x_bse, v_bse
  float iser = var0, isei = 0.f;                           // ise_dsc

  const int cbase = (lane >> 3) * 4;   // 4 constellation points per lane

  for (int it = 0; it < iters; ++it) {
    // ---- BSO: t = Hty - HtH_off @ x_dsc ; s2 = HtH_off_sqr @ v_dsc
    float tre = tyre, tim = tyim, s2r = 0.f, s2i = 0.f;
#pragma unroll
    for (int c = 0; c < 8; ++c) {
      const float xr = __shfl(xdr, c, 32), xi = __shfl(xdi, c, 32);
      const float vr = __shfl(vdr, c, 32), vi = __shfl(vdi, c, 32);
      tre -= hr[c] * xr - hi[c] * xi;
      tim -= hr[c] * xi + hi[c] * xr;
      s2r += sr[c] * vr - si[c] * vi;
      s2i += sr[c] * vi + si[c] * vr;
    }
    const float xsr = mre * tre - mim * tim;               // x_bso
    const float xsi = mre * tim + mim * tre;
    const float vbr = No * mre + s2r * m2r - s2i * m2i;    // v_bso (complex)
    const float vbi = No * mim + s2r * m2i + s2i * m2r;
    const float vden = vbr * vbr + vbi * vbi;
    const float wre = -0.5f * vbr / vden;                  // -1/(2 v_bso)
    const float wim =  0.5f * vbi / vden;

    // ---- BSE: complex "power", argmax by Re, complex-exp pdf
    float prr[4], pri[4];
    float bmr = -1e30f, bmi = 0.f;
#pragma unroll
    for (int t = 0; t < 4; ++t) {
      const float2 P = cpt(cbase + t);
      const float dr = P.x - xsr, di = P.y - xsi;
      const float d2 = dr * dr + di * di;
      prr[t] = wre * d2; pri[t] = wim * d2;
      if (prr[t] > bmr) { bmr = prr[t]; bmi = pri[t]; }
    }
#pragma unroll
    for (int m = 8; m <= 16; m <<= 1) {      // max over lanes {k,k+8,k+16,k+24}
      const float orr = __shfl_xor(bmr, m, 32);
      const float ori = __shfl_xor(bmi, m, 32);
      if (orr > bmr) { bmr = orr; bmi = ori; }
    }
    float Sre = 0.f, Sim = 0.f, Sxr = 0.f, Sxi = 0.f;
    float fr[4], fi[4];
#pragma unroll
    for (int t = 0; t < 4; ++t) {
      const float er = __expf(prr[t] - bmr);
      float sa, ca;
      __sincosf(pri[t] - bmi, &sa, &ca);
      fr[t] = er * ca; fi[t] = er * sa;      // complex exp
      const float2 P = cpt(cbase + t);
      Sre += fr[t]; Sim += fi[t];
      Sxr += fr[t] * P.x - fi[t] * P.y;
      Sxi += fr[t] * P.y + fi[t] * P.x;
    }
#pragma unroll
    for (int m = 8; m <= 16; m <<= 1) {
      Sre += __shfl_xor(Sre, m, 32); Sim += __shfl_xor(Sim, m, 32);
      Sxr += __shfl_xor(Sxr, m, 32); Sxi += __shfl_xor(Sxi, m, 32);
    }
    const float sden = Sre * Sre + Sim * Sim;
    xbr = (Sxr * Sre + Sxi * Sim) / sden;    // x_bse = Sx / S
    xbi = (Sxi * Sre - Sxr * Sim) / sden;

    float Svr = 0.f, Svi = 0.f;
#pragma unroll
    for (int t = 0; t < 4; ++t) {
      const float2 P = cpt(cbase + t);
      const float dr = P.x - xbr, di = P.y - xbi;
      const float d2 = dr * dr + di * di;
      Svr += d2 * fr[t]; Svi += d2 * fi[t];
    }
#pragma unroll
    for (int m = 8; m <= 16; m <<= 1) {
      Svr += __shfl_xor(Svr, m, 32); Svi += __shfl_xor(Svi, m, 32);
    }
    vor = (Svr * Sre + Svi * Sim) / sden;    // v_bse = Sv / S
    voi = (Svi * Sre - Svr * Sim) / sden;

    // ---- DSC instantaneous square error: (mrc*(Hty - HtH @ x_bse))^2
    float rre = tyre, rim = tyim;
#pragma unroll
    for (int c = 0; c < 8; ++c) {
      const float xr = __shfl(xbr, c, 32), xi = __shfl(xbi, c, 32);
      const float are = (c == k) ? dre : hr[c];
      const float aim = (c == k) ? dim : hi[c];
      rre -= are * xr - aim * xi;
      rim -= are * xi + aim * xr;
    }
    const float zr = mre * rre - mim * rim;
    const float zi = mre * rim + mim * rre;
    iser = zr * zr - zi * zi;                // elementwise complex square
    isei = 2.f * zr * zi;

    xdr = xbr; xdi = xbi; vdr = vor; vdi = voi;
  }

  // ================= outputs (lanes 0..7 only) =================
  if ((lane >> 3) == 0) {
    const size_t idx = (size_t)p * 8 + k;
    const size_t N   = (size_t)NPROB * 8;
    out[idx * 2]           = xbr;  out[idx * 2 + 1]           = xbi;
    out[(N + idx) * 2]     = vor;  out[(N + idx) * 2 + 1]     = voi;
    out[(2 * N + idx) * 2] = iser; out[(2 * N + idx) * 2 + 1] = isei;
  }
}

extern "C" void kernel_launch(void* const* d_in, const int* in_sizes, int n_in,
                              void* d_out, int out_size, void* d_ws, size_t ws_size,
                              hipStream_t stream) {
  // setup_inputs order: y,h,prior,no,it,v_dsc_prev,x_bse_prev,v_bse_prev,
  //                     ise_dsc_prev,test,x_dsc_prev,internal_it
  const float* y   = (const float*)d_in[0];
  const float* h   = (const float*)d_in[1];
  const float* pr  = (const float*)d_in[2];
  const float* no  = (const float*)d_in[3];
  const int*   iit = (const int*)d_in[11];

  dim3 grid(NPROB / 8), block(256);      // one wave32 per problem, 8 waves/block
  bpic_wmma_kernel<<<grid, block, 0, stream>>>(y, h, pr, no, iit, (float*)d_out);
}